// GTN_6846177870061
// MI455X (gfx1250) — compile-verified
//
#include <hip/hip_runtime.h>

// Problem constants (from reference)
#define NU_ 50000
#define NI_ 100000
#define NN_ 150000     // NU + NI
#define DD_ 64
#define EE_ 1250000
#define BB_ 16384
static constexpr float ALPHA_ = 0.1f;

typedef float v2f __attribute__((ext_vector_type(2)));
typedef float v8f __attribute__((ext_vector_type(8)));

// ---------------------------------------------------------------------------
// x = concat(user_emb, item_emb), as float4 (16 float4 per node row)
// ---------------------------------------------------------------------------
__global__ void lgcn_init_x0(float4* __restrict__ x,
                             const float4* __restrict__ ue,
                             const float4* __restrict__ ie) {
    int i = blockIdx.x * blockDim.x + threadIdx.x;   // float4 index
    if (i >= NN_ * 16) return;
    int node = i >> 4;
    x[i] = (node < NU_) ? ue[i] : ie[i - NU_ * 16];
}

// ---------------------------------------------------------------------------
// zero the aggregation buffer
// ---------------------------------------------------------------------------
__global__ void lgcn_zero(float4* __restrict__ p) {
    int i = blockIdx.x * blockDim.x + threadIdx.x;
    if (i >= NN_ * 16) return;
    p[i] = make_float4(0.f, 0.f, 0.f, 0.f);
}

// ---------------------------------------------------------------------------
// agg[row[e]] += val[e] * x[col[e]]  -- 16 threads per edge, float4 per thread,
// hardware fp32 atomics into the L2-resident accumulator.
// ---------------------------------------------------------------------------
__global__ void lgcn_scatter(const float* __restrict__ x,
                             float* __restrict__ agg,
                             const int* __restrict__ erow,
                             const int* __restrict__ ecol,
                             const float* __restrict__ eval_) {
    int t = blockIdx.x * blockDim.x + threadIdx.x;
    int e = t >> 4;
    if (e >= EE_) return;
    int d = (t & 15) << 2;

    // stream the edge arrays ahead (global_prefetch_b8, speculative -> OOB safe)
    if ((t & 1023) == 0) {
        __builtin_prefetch(erow + e + 8192, 0, 1);
        __builtin_prefetch(ecol + e + 8192, 0, 1);
        __builtin_prefetch(eval_ + e + 8192, 0, 1);
    }

    int r = erow[e];
    int c = ecol[e];
    float w = eval_[e];

    const float4 xv = *reinterpret_cast<const float4*>(x + (size_t)c * DD_ + d);
    float* dst = agg + (size_t)r * DD_ + d;
    unsafeAtomicAdd(dst + 0, w * xv.x);
    unsafeAtomicAdd(dst + 1, w * xv.y);
    unsafeAtomicAdd(dst + 2, w * xv.z);
    unsafeAtomicAdd(dst + 3, w * xv.w);
}

// ---------------------------------------------------------------------------
// x_new = (1-ALPHA)*agg + ALPHA*x0, in-place on agg; x0 read from inputs.
// ---------------------------------------------------------------------------
__global__ void lgcn_combine(float4* __restrict__ xn,
                             const float4* __restrict__ ue,
                             const float4* __restrict__ ie) {
    int i = blockIdx.x * blockDim.x + threadIdx.x;
    if (i >= NN_ * 16) return;
    int node = i >> 4;
    float4 z = (node < NU_) ? ue[i] : ie[i - NU_ * 16];
    float4 a = xn[i];
    float4 r;
    r.x = fmaf(1.0f - ALPHA_, a.x, ALPHA_ * z.x);
    r.y = fmaf(1.0f - ALPHA_, a.y, ALPHA_ * z.y);
    r.z = fmaf(1.0f - ALPHA_, a.z, ALPHA_ * z.z);
    r.w = fmaf(1.0f - ALPHA_, a.w, ALPHA_ * z.w);
    xn[i] = r;
}

// ---------------------------------------------------------------------------
// gamma[b] = dot(X[users[b]], X[NU+items[b]]) via V_WMMA_F32_16X16X4_F32.
// One wave handles 16 pairs: S = U (16x64) @ I^T (64x16) accumulated in 16
// k-steps of 4; diagonal of the 16x16 result is the answer.
//
// A (16x4 f32, 2 VGPRs): lanes 0-15 = M rows, v0=K0 v1=K1; lanes 16-31 K2/K3.
// B (4x16 f32, 2 VGPRs): lanes 0-15 = N cols, v0=K0 v1=K1; lanes 16-31 K2/K3.
// C/D (16x16 f32, 8 VGPRs): vgpr r, lanes 0-15 -> M=r, N=lane;
//                           lanes 16-31 -> M=r+8, N=lane-16.
// ---------------------------------------------------------------------------
__global__ void lgcn_gamma_wmma(const float* __restrict__ X,
                                const int* __restrict__ users,
                                const int* __restrict__ items,
                                float* __restrict__ out) {
    __shared__ float smem[8][16 * 16];

    const int lane = threadIdx.x & 31;
    const int wave = threadIdx.x >> 5;
    const int tile = blockIdx.x * 8 + wave;     // 16 pairs per tile
    const int half = lane >> 4;                 // K sub-half within a 4-chunk
    const int idx  = lane & 15;                 // M row (for A) / N col (for B)

    const int b = tile * 16 + idx;
    const size_t urow = (size_t)users[b] * DD_;
    const size_t irow = (size_t)(NU_ + items[b]) * DD_;

    v8f c = {};
#pragma unroll
    for (int ka = 0; ka < DD_; ka += 4) {
        v2f a, bb;
        a.x  = X[urow + ka + 2 * half + 0];
        a.y  = X[urow + ka + 2 * half + 1];
        bb.x = X[irow + ka + 2 * half + 0];
        bb.y = X[irow + ka + 2 * half + 1];
        // 8 args: (neg_a, A, neg_b, B, c_mod, C, reuse_a, reuse_b)
        c = __builtin_amdgcn_wmma_f32_16x16x4_f32(
                false, a, false, bb, (short)0, c, false, false);
    }

    // spill the 16x16 tile to LDS, then lanes 0-15 read the diagonal
#pragma unroll
    for (int r = 0; r < 8; ++r) {
        smem[wave][(r + 8 * half) * 16 + idx] = c[r];
    }
    __syncthreads();
    if (lane < 16) {
        out[tile * 16 + lane] = smem[wave][lane * 16 + lane];
    }
}

// ---------------------------------------------------------------------------
// Host-side launch sequence (graph-capture safe: only kernel launches)
// ---------------------------------------------------------------------------
extern "C" void kernel_launch(void* const* d_in, const int* in_sizes, int n_in,
                              void* d_out, int out_size, void* d_ws, size_t ws_size,
                              hipStream_t stream) {
    const float* user_emb = (const float*)d_in[0];   // NU*64 f32
    const float* item_emb = (const float*)d_in[1];   // NI*64 f32
    const float* edge_val = (const float*)d_in[2];   // E f32
    const int*   users    = (const int*)d_in[3];     // B i32
    const int*   items    = (const int*)d_in[4];     // B i32
    const int*   erow     = (const int*)d_in[5];     // E i32
    const int*   ecol     = (const int*)d_in[6];     // E i32

    float* P = (float*)d_ws;                 // NN*64 floats (38.4 MB)
    float* Q = P + (size_t)NN_ * DD_;        // NN*64 floats (38.4 MB)

    const int n4 = NN_ * 16;                 // float4 elements per buffer
    dim3 blk(256);
    dim3 grd_n((n4 + 255) / 256);
    dim3 grd_e(((long long)EE_ * 16 + 255) / 256);

    lgcn_init_x0<<<grd_n, blk, 0, stream>>>(
        (float4*)P, (const float4*)user_emb, (const float4*)item_emb);

    float* cur = P;
    float* nxt = Q;
    for (int k = 0; k < 3; ++k) {
        lgcn_zero<<<grd_n, blk, 0, stream>>>((float4*)nxt);
        lgcn_scatter<<<grd_e, blk, 0, stream>>>(cur, nxt, erow, ecol, edge_val);
        lgcn_combine<<<grd_n, blk, 0, stream>>>(
            (float4*)nxt, (const float4*)user_emb, (const float4*)item_emb);
        float* t = cur; cur = nxt; nxt = t;
    }

    // B=16384 pairs -> 1024 tiles of 16; 8 waves (tiles) per 256-thread block
    lgcn_gamma_wmma<<<BB_ / 128, 256, 0, stream>>>(cur, users, items, (float*)d_out);
}